// CrossViewFusionModule_2035814498647
// MI455X (gfx1250) — compile-verified
//
#include <hip/hip_runtime.h>
#include <hip/hip_bf16.h>
#include <math.h>

// Problem constants (from reference setup_inputs)
#define B_  32
#define D_  512
#define WH_ 4096

typedef __attribute__((ext_vector_type(2))) float v2f;
typedef __attribute__((ext_vector_type(4))) float v4f;
typedef __attribute__((ext_vector_type(8))) float v8f;

// ---------------------------------------------------------------------------
// Pass 1: per-block = (batch b, tile of 128 pixels).  8 waves x 16 pixels.
//  - cooperatively L2-normalize q[b,:] into LDS
//  - per wave: accumulate score via V_WMMA_F32_16X16X4_F32 over D in chunks
//    of 4, and per-pixel sum-of-squares of value via VALU on the same loads.
// value read exactly once (256 MiB total across grid) -> bandwidth bound.
// ---------------------------------------------------------------------------
__global__ __launch_bounds__(256)
void cvf_pass1_score(const float* __restrict__ gq,
                     const float* __restrict__ value,
                     float* __restrict__ score,
                     float* __restrict__ inv_vn) {
    const int b    = blockIdx.y;
    const int tile = blockIdx.x;            // 32 tiles of 128 pixels
    const int tid  = threadIdx.x;

    __shared__ float qn[D_];
    __shared__ float red[256];

    // ---- normalize q[b,:] into LDS (redundant per block; 2 KB, negligible)
    float s0 = 0.f;
    #pragma unroll
    for (int d = tid; d < D_; d += 256) {
        float x = gq[b * D_ + d];
        qn[d] = x;
        s0 += x * x;
    }
    red[tid] = s0;
    __syncthreads();
    #pragma unroll
    for (int off = 128; off > 0; off >>= 1) {
        if (tid < off) red[tid] += red[tid + off];
        __syncthreads();
    }
    const float qinv = 1.0f / fmaxf(sqrtf(red[0]), 1e-12f);
    #pragma unroll
    for (int d = tid; d < D_; d += 256) qn[d] *= qinv;
    __syncthreads();

    // ---- wave-level WMMA GEMV
    const int wave = tid >> 5;              // 0..7
    const int lane = tid & 31;
    const int half = lane >> 4;             // 0: K=0,1   1: K=2,3  (A/B layout)
    const int l16  = lane & 15;             // pixel within 16-wide tile (N)
    const int pix  = tile * 128 + wave * 16 + l16;

    const float* __restrict__ V = value + (size_t)b * D_ * WH_;

    v8f  c  = {};                           // 16x16 f32 accumulator (row0 = score)
    float ss = 0.f;                         // per-lane partial sum-of-squares

    for (int d0 = 0; d0 < D_; d0 += 4) {
        const int kb = d0 + 2 * half;
        // B matrix 4x16: lanes 0-15 hold rows K=0,1; lanes 16-31 rows K=2,3
        const float bx = V[(size_t)kb * WH_ + pix];
        const float by = V[(size_t)(kb + 1) * WH_ + pix];
        // prefetch two chunks ahead (speculative; dropped if OOB)
        __builtin_prefetch(&V[(size_t)(kb + 8) * WH_ + pix], 0, 1);

        ss += bx * bx + by * by;

        // A matrix 16x4 with every row = q-hat chunk (uniform across M)
        v2f a;  a.x = qn[kb]; a.y = qn[kb + 1];
        v2f bv; bv.x = bx;    bv.y = by;
        c = __builtin_amdgcn_wmma_f32_16x16x4_f32(
                /*neg_a=*/false, a, /*neg_b=*/false, bv,
                /*c_mod=*/(short)0, c, /*reuse_a=*/false, /*reuse_b=*/false);
    }

    // combine sumsq halves: lane L holds d%4 in {0,1}, lane L+16 holds {2,3}
    ss += __shfl_xor(ss, 16, 32);
    const float vinv = 1.0f / fmaxf(sqrtf(ss), 1e-12f);

    // c[0]: lanes 0-15 -> D[0][N]  (score for this wave's 16 pixels);
    // upper half holds the identical replicated row M=8.
    const float sc = c[0] * vinv;
    if (lane < 16) {
        score[b * WH_ + pix]  = sc;
        inv_vn[b * WH_ + pix] = vinv;
    }
}

// ---------------------------------------------------------------------------
// Pass 2: one block per batch: min/max over 4096 scores, write attn to d_out
// and fused scale = attn * inv_vnorm for pass 3. (~2 MB traffic total)
// ---------------------------------------------------------------------------
__global__ __launch_bounds__(256)
void cvf_pass2_attn(const float* __restrict__ score,
                    const float* __restrict__ inv_vn,
                    float* __restrict__ attn_out,
                    float* __restrict__ scale) {
    const int b   = blockIdx.x;
    const int tid = threadIdx.x;
    __shared__ float smn[256], smx[256];

    float mn =  INFINITY, mx = -INFINITY;
    #pragma unroll
    for (int p = tid; p < WH_; p += 256) {
        const float s = score[b * WH_ + p];
        mn = fminf(mn, s);
        mx = fmaxf(mx, s);
    }
    smn[tid] = mn; smx[tid] = mx;
    __syncthreads();
    #pragma unroll
    for (int off = 128; off > 0; off >>= 1) {
        if (tid < off) {
            smn[tid] = fminf(smn[tid], smn[tid + off]);
            smx[tid] = fmaxf(smx[tid], smx[tid + off]);
        }
        __syncthreads();
    }
    mn = smn[0]; mx = smx[0];
    const float inv = 1.0f / (mx - mn);

    #pragma unroll
    for (int p = tid; p < WH_; p += 256) {
        const float a = (score[b * WH_ + p] - mn) * inv;
        attn_out[b * WH_ + p] = a;
        scale[b * WH_ + p]    = a * inv_vn[b * WH_ + p];
    }
}

// ---------------------------------------------------------------------------
// Pass 3: context[b,d,p] = scale[b,p] * value[b,d,p].  Pure stream:
// float4 loads; non-temporal stores so the 256 MiB output does not evict
// value lines still resident in the 192 MB L2 from pass 1.
// ---------------------------------------------------------------------------
__global__ __launch_bounds__(256)
void cvf_pass3_context(const float* __restrict__ value,
                       const float* __restrict__ scale,
                       float* __restrict__ ctx) {
    const size_t idx = (size_t)blockIdx.x * 256 + threadIdx.x; // float4 index
    const v4f v = ((const v4f*)value)[idx];

    const int row = (int)(idx >> 10);       // b*512 + d   (1024 float4 per row)
    const int b   = row >> 9;
    const int p4  = (int)(idx & 1023);
    const v4f s = ((const v4f*)scale)[(size_t)b * (WH_ / 4) + p4];

    v4f o;
    o.x = v.x * s.x; o.y = v.y * s.y; o.z = v.z * s.z; o.w = v.w * s.w;
    __builtin_nontemporal_store(o, ((v4f*)ctx) + idx);
}

// ---------------------------------------------------------------------------
extern "C" void kernel_launch(void* const* d_in, const int* in_sizes, int n_in,
                              void* d_out, int out_size, void* d_ws, size_t ws_size,
                              hipStream_t stream) {
    const float* gq    = (const float*)d_in[0];   // [32, 512]
    const float* value = (const float*)d_in[1];   // [32, 512, 64, 64]

    float* out  = (float*)d_out;
    float* ctx  = out;                                  // [B, D, W, H]
    float* attn = out + (size_t)B_ * D_ * WH_;          // [B, 1, W, H]

    float* score  = (float*)d_ws;                       // [B, WH]
    float* inv_vn = score  + B_ * WH_;                  // [B, WH]
    float* scale  = inv_vn + B_ * WH_;                  // [B, WH]

    // Pass 1: 32 pixel-tiles x 32 batches, 256 threads (8 waves x 16 pixels)
    cvf_pass1_score<<<dim3(WH_ / 128, B_), 256, 0, stream>>>(gq, value, score, inv_vn);

    // Pass 2: per-batch min/max + attn + fused scale
    cvf_pass2_attn<<<B_, 256, 0, stream>>>(score, inv_vn, attn, scale);

    // Pass 3: stream-scale value -> context (16,777,216 float4 elements)
    const int n4 = (B_ * D_ * WH_) / 4;
    cvf_pass3_context<<<n4 / 256, 256, 0, stream>>>(value, scale, ctx);
}